// MultivariateKDE_53506702573582
// MI455X (gfx1250) — compile-verified
//
#include <hip/hip_runtime.h>

typedef __attribute__((ext_vector_type(2))) float v2f;
typedef __attribute__((ext_vector_type(8))) float v8f;

#define KDE_D 128
#define KDE_P (KDE_D / 2)     // 64 d-pairs
#define KTILES_PER_WAVE 32    // 512 train columns per wave

// ---------------------------------------------------------------------------
// Kernel 1: Scott bandwidth per dimension + Z normalizer.
// 1 block, 128 threads (one per dimension). Double accumulation for the
// 16384-element mean/var reductions (trivial cost, robust).
// ---------------------------------------------------------------------------
__global__ void kde_bw_kernel(const float* __restrict__ train, int n,
                              float* __restrict__ bw, float* __restrict__ inv_bw,
                              float* __restrict__ scal) {
    const int d = threadIdx.x;  // 0..127
    double s = 0.0, s2 = 0.0;
    for (int i = 0; i < n; ++i) {
        float x = train[i * KDE_D + d];  // coalesced across the 128 threads
        s += (double)x;
        s2 += (double)x * (double)x;
    }
    double var = (s2 - s * s / (double)n) / (double)(n - 1);
    if (var < 0.0) var = 0.0;
    float stdv = (float)sqrt(var);
    stdv = fmaxf(stdv, 0.01f);
    // n^(-1/(4+D)) with D=128 -> exponent -1/132
    float scott = (float)exp(-log((double)n) / 132.0);
    float b = 1.06f * stdv * scott;
    b = fminf(b, 0.49f);
    bw[d] = b;
    inv_bw[d] = 1.0f / b;

    __shared__ float red[KDE_D];
    red[d] = logf(b);
    __syncthreads();
    for (int off = KDE_D / 2; off > 0; off >>= 1) {
        if (d < off) red[d] += red[d + off];
        __syncthreads();
    }
    if (d == 0) {
        // Z = 0.5*D*log(2*pi) + sum(log bw) + log(n)
        float Z = 0.5f * (float)KDE_D * logf(6.28318530717958647692f)
                + red[0] + logf((float)n);
        scal[0] = Z;
        scal[1] = 1.0f / Z;
    }
}

// ---------------------------------------------------------------------------
// Kernel 2: scale by 1/bw, transpose, and pack adjacent d-rows into float2:
//   dstP[p*n + q] = { src[q][2p]/bw[2p], src[q][2p+1]/bw[2p+1] }
// q varies fastest across threads -> coalesced 8B writes.
// ---------------------------------------------------------------------------
__global__ void kde_pack_kernel(const float* __restrict__ src,
                                const float* __restrict__ inv_bw,
                                float* __restrict__ dstP, int n) {
    int i = blockIdx.x * blockDim.x + threadIdx.x;
    if (i >= KDE_P * n) return;
    int q = i % n;
    int p = i / n;
    float2 v;
    v.x = src[q * KDE_D + 2 * p + 0] * inv_bw[2 * p + 0];
    v.y = src[q * KDE_D + 2 * p + 1] * inv_bw[2 * p + 1];
    ((float2*)dstP)[p * n + q] = v;
}

// ---------------------------------------------------------------------------
// Kernel 3: row squared-norms from the packed layout (coalesced b64 reads).
// ---------------------------------------------------------------------------
__global__ void kde_norms_kernel(const float* __restrict__ srcP, int n,
                                 float* __restrict__ out) {
    int q = blockIdx.x * blockDim.x + threadIdx.x;
    if (q >= n) return;
    float acc = 0.0f;
    const float2* s2 = (const float2*)srcP;
    for (int p = 0; p < KDE_P; ++p) {
        float2 v = s2[p * n + q];
        acc = fmaf(v.x, v.x, acc);
        acc = fmaf(v.y, v.y, acc);
    }
    out[q] = acc;
}

// ---------------------------------------------------------------------------
// Kernel 4: WMMA distance GEMM + fused exp + per-row partial sums.
// Block = 256 threads = 8 waves; wave w owns q-tile (blockIdx.x*8 + w) and
// k-group blockIdx.y (512 train columns = 32 tiles of 16).
//
// V_WMMA_F32_16X16X4_F32 fragment layout (ISA 7.12.2):
//   A (16x4):  lanes 0-15 rows M=lane, VGPR0=K0 VGPR1=K1; lanes 16-31 K2/K3
//   B (4x16):  lanes 0-15 cols N=lane, VGPR0=K0 VGPR1=K1; lanes 16-31 K2/K3
//   C/D:       VGPR r, lane l -> (M = r + 8*(l>>4), N = l&15)
// With the pair-packed operands, each fragment half is ONE global_load_b64:
//   pair index p = 2*s4 + (lane>>4)  ->  d = 4*s4 + 2*hi, d+1.
// B tiles are double-buffered (b0/b1) so the 32-deep WMMA accumulation chain
// and the exp stage overlap the next tile's loads.
// EXEC stays all ones: exact grid coverage, only uniform scalar branches.
// ---------------------------------------------------------------------------
__global__ __launch_bounds__(256)
void kde_wmma_kernel(const float* __restrict__ tP, const float* __restrict__ trP,
                     const float* __restrict__ qn, const float* __restrict__ kn,
                     const float* __restrict__ scal, float* __restrict__ partial,
                     int NQ, int NK, int nkg) {
    const int lane = threadIdx.x & 31;
    const int wave = threadIdx.x >> 5;
    const int qt   = blockIdx.x * 8 + wave;
    const int kg   = blockIdx.y;
    const int q0   = qt * 16;
    const int hi   = lane >> 4;       // which half of the wave (K sub-offset)
    const int lo   = lane & 15;       // row (A) / col (B,D) index
    const float invZ = scal[1];

    const float2* tP2  = (const float2*)tP;
    const float2* trP2 = (const float2*)trP;

    // Hoist the full-D A fragment for this q-tile (reused for all 32 k-tiles).
    v2f aF[32];
#pragma unroll
    for (int s4 = 0; s4 < 32; ++s4) {
        float2 t = tP2[(2 * s4 + hi) * NQ + q0 + lo];
        aF[s4].x = t.x;
        aF[s4].y = t.y;
    }
    // qn for the 8 output rows this lane's accumulator elements belong to.
    float qrow[8];
#pragma unroll
    for (int r = 0; r < 8; ++r) qrow[r] = qn[q0 + 8 * hi + r];

    float racc[8] = {};
    const int kstart = kg * (KTILES_PER_WAVE * 16);

    v2f b0[32], b1[32];

    auto loadB = [&](v2f (&buf)[32], int kb) {
#pragma unroll
        for (int s4 = 0; s4 < 32; ++s4) {
            float2 t = trP2[(2 * s4 + hi) * NK + kb + lo];
            buf[s4].x = t.x;
            buf[s4].y = t.y;
        }
    };
    auto processB = [&](const v2f (&buf)[32], int kb) {
        v8f acc = {};
#pragma unroll
        for (int s4 = 0; s4 < 32; ++s4) {
            acc = __builtin_amdgcn_wmma_f32_16x16x4_f32(
                false, aF[s4], false, buf[s4], (short)0, acc, false, false);
        }
        const float knv = kn[kb + lo];
#pragma unroll
        for (int r = 0; r < 8; ++r) {
            float sq = fmaxf(qrow[r] + knv - 2.0f * acc[r], 0.0f);
            // exp((-0.5*sq - Z)/Z) = exp(-0.5*sq*invZ - 1)
            racc[r] += __expf(fmaf(-0.5f * sq, invZ, -1.0f));
        }
    };

    loadB(b0, kstart);
    for (int ktp = 0; ktp < KTILES_PER_WAVE / 2; ++ktp) {
        const int kb0 = kstart + ktp * 32;
        loadB(b1, kb0 + 16);                                   // prefetch odd tile
        processB(b0, kb0);                                     // compute even tile
        if (ktp + 1 < KTILES_PER_WAVE / 2) loadB(b0, kb0 + 32); // prefetch next even
        processB(b1, kb0 + 16);                                // compute odd tile
    }

    // Sum each row across its 16 lanes (halves are independent).
#pragma unroll
    for (int r = 0; r < 8; ++r) {
        float v = racc[r];
        v += __shfl_xor(v, 1, 32);
        v += __shfl_xor(v, 2, 32);
        v += __shfl_xor(v, 4, 32);
        v += __shfl_xor(v, 8, 32);
        racc[r] = v;
    }
    if (lo == 0) {
#pragma unroll
        for (int r = 0; r < 8; ++r) {
            int row = q0 + 8 * hi + r;
            partial[row * nkg + kg] = racc[r];
        }
    }
}

// ---------------------------------------------------------------------------
// Kernel 5: deterministic final reduce over k-groups.
// ---------------------------------------------------------------------------
__global__ void kde_reduce_kernel(const float* __restrict__ partial,
                                  float* __restrict__ out, int NQ, int nkg) {
    int q = blockIdx.x * blockDim.x + threadIdx.x;
    if (q >= NQ) return;
    float acc = 0.0f;
    for (int g = 0; g < nkg; ++g) acc += partial[q * nkg + g];
    out[q] = acc;
}

extern "C" void kernel_launch(void* const* d_in, const int* in_sizes, int n_in,
                              void* d_out, int out_size, void* d_ws, size_t ws_size,
                              hipStream_t stream) {
    const float* test  = (const float*)d_in[0];
    const float* train = (const float*)d_in[1];
    float* out = (float*)d_out;

    const int NQ = in_sizes[0] / KDE_D;   // 4096
    const int NK = in_sizes[1] / KDE_D;   // 16384
    const int nkg = NK / (KTILES_PER_WAVE * 16);  // 32

    // Workspace carve-up (floats); all float2 regions are 8B-aligned.
    float* ws      = (float*)d_ws;
    float* bw      = ws;                  // 128
    float* inv_bw  = ws + KDE_D;          // 128
    float* scal    = ws + 2 * KDE_D;      // 2 (Z, 1/Z); padded region
    float* qn      = ws + 3 * KDE_D;      // NQ
    float* kn      = qn + NQ;             // NK
    float* tP      = kn + NK;             // D*NQ  (pair-packed)
    float* trP     = tP + KDE_D * NQ;     // D*NK  (pair-packed)
    float* partial = trP + KDE_D * NK;    // NQ*nkg

    kde_bw_kernel<<<1, KDE_D, 0, stream>>>(train, NK, bw, inv_bw, scal);

    int tot_t  = KDE_P * NQ;
    int tot_tr = KDE_P * NK;
    kde_pack_kernel<<<(tot_t + 255) / 256, 256, 0, stream>>>(test, inv_bw, tP, NQ);
    kde_pack_kernel<<<(tot_tr + 255) / 256, 256, 0, stream>>>(train, inv_bw, trP, NK);

    kde_norms_kernel<<<(NQ + 255) / 256, 256, 0, stream>>>(tP, NQ, qn);
    kde_norms_kernel<<<(NK + 255) / 256, 256, 0, stream>>>(trP, NK, kn);

    dim3 grid(NQ / 128, nkg);   // 8 q-tiles of 16 rows per block, nkg k-groups
    kde_wmma_kernel<<<grid, 256, 0, stream>>>(tP, trP, qn, kn, scal, partial, NQ, NK, nkg);

    kde_reduce_kernel<<<(NQ + 255) / 256, 256, 0, stream>>>(partial, out, NQ, nkg);
}